// MPSGNN_Original_40535901339974
// MI455X (gfx1250) — compile-verified
//
#include <hip/hip_runtime.h>

// The whole reference collapses: h["drivers"] is never updated by any hop
// (dst is always races/results), and both metapaths read h["drivers"] at the
// end. All edge gather/scatter work is dead code. With no nonlinearity
// between out_proj and fc1, the two linears fold into a single 64x64 matrix:
//   Wcomb = wproj[0] @ wfc1[0:64]  + wproj[1] @ wfc1[64:128]
//   bcomb = bproj[0] @ wfc1[0:64]  + bproj[1] @ wfc1[64:128] + bfc1
//   out[i] = relu(x_drivers[i] @ Wcomb + bcomb) . wfc2 + bfc2
// Memory floor: 128 MB of X reads -> ~5.6us @ 23.3 TB/s. fp32 WMMA
// (v_wmma_f32_16x16x4_f32) keeps full precision; AI = 32 FLOP/B puts this at
// the fp32-WMMA roofline knee, so the inner loop is kept lean: B fragments
// are pre-packed in LDS so each one is a single aligned ds_load_b64 (no VGPR
// repacking movs), with a bank-disjoint pitch for the two 16-lane halves.

typedef __attribute__((ext_vector_type(2))) float v2f;
typedef __attribute__((ext_vector_type(4))) float v4f;
typedef __attribute__((ext_vector_type(8))) float v8f;

#define HDIM   64
#define ROWS   128           // rows per block (8 waves x 16 rows)
#define XPITCH 68            // X tile pitch: even (8B v2f), mult of 4 (16B
                             // v4f), 68%64=4 -> conflict-free A-frag reads
#define WPITCH 160           // packed-W k-pair row pitch (floats): 160%64=32
                             // -> the two 16-lane halves (k-pair rows r, r+1)
                             // hit disjoint bank halves on B-frag b64 reads

// ---------------------------------------------------------------------------
// Kernel 1: fold wproj/bproj + wfc1/bfc1 into Wcomb (64x64) and bcomb (64).
// ws layout: ws[0..4095] = Wcomb row-major, ws[4096..4159] = bcomb.
// ---------------------------------------------------------------------------
__global__ __launch_bounds__(256) void build_combined_weights(
    const float* __restrict__ wproj,   // [2,64,64]
    const float* __restrict__ bproj,   // [2,64]
    const float* __restrict__ wfc1,    // [128,64]
    const float* __restrict__ bfc1,    // [64]
    float* __restrict__ ws)
{
    const int tid = threadIdx.x;
    for (int e = 0; e < 16; ++e) {
        const int idx = tid * 16 + e;      // 0..4095
        const int i = idx >> 6;
        const int j = idx & 63;
        float acc = 0.0f;
        #pragma unroll 4
        for (int k = 0; k < HDIM; ++k) {
            acc += wproj[i * 64 + k]          * wfc1[k * 64 + j];
            acc += wproj[4096 + i * 64 + k]   * wfc1[(64 + k) * 64 + j];
        }
        ws[idx] = acc;
    }
    if (tid < 64) {
        const int j = tid;
        float acc = bfc1[j];
        #pragma unroll 4
        for (int k = 0; k < HDIM; ++k) {
            acc += bproj[k]      * wfc1[k * 64 + j];
            acc += bproj[64 + k] * wfc1[(64 + k) * 64 + j];
        }
        ws[4096 + j] = acc;
    }
}

// ---------------------------------------------------------------------------
// Kernel 2: out = relu(X @ Wcomb + bcomb) . wfc2 + bfc2, via fp32 WMMA.
// 8 waves/block, 16 rows/wave, 4 col-tiles of 16 -> 4x v8f accumulators.
// ---------------------------------------------------------------------------
__global__ __launch_bounds__(256) void mpsgnn_fused_gemm(
    const float* __restrict__ X,       // [N,64]
    const float* __restrict__ Wcomb,   // [64,64] (in ws)
    const float* __restrict__ bcomb,   // [64]    (in ws)
    const float* __restrict__ wfc2,    // [64]  (wfc2 is [64,1])
    const float* __restrict__ bfc2,    // [1]
    float* __restrict__ out,           // [N]
    int n)
{
    // Packed W: element (k, c) lives at sWp[(k>>1)*WPITCH + c*2 + (k&1)],
    // so a B fragment (K={kk,kk+1}, col c) is ONE aligned v2f/b64 LDS load.
    __shared__ float sWp[32 * WPITCH];     // 20 KB
    __shared__ float sX[ROWS * XPITCH];    // 34 KB
    __shared__ float sB[HDIM];
    __shared__ float sW2[HDIM];

    const int tid     = threadIdx.x;
    const int rowbase = blockIdx.x * ROWS;

    // Stage Wcomb with K-pair interleave (coalesced b128 reads, strided
    // b32 LDS writes; one-time cost vs 16-iteration inner loop).
    {
        #pragma unroll
        for (int p = 0; p < 4; ++p) {
            const int idx = tid + p * 256;         // float4 index, 0..1023
            const int k   = idx >> 4;              // row of Wcomb
            const int c0  = (idx & 15) * 4;        // first of 4 columns
            const v4f v = ((const v4f*)Wcomb)[idx];
            float* dst = sWp + (k >> 1) * WPITCH + (k & 1);
            dst[(c0 + 0) * 2] = v.x;
            dst[(c0 + 1) * 2] = v.y;
            dst[(c0 + 2) * 2] = v.z;
            dst[(c0 + 3) * 2] = v.w;
        }
    }
    if (tid < 64) { sB[tid] = bcomb[tid]; sW2[tid] = wfc2[tid]; }

    // Stage X tile (128 rows x 64 cols), zero-pad out-of-range rows.
    #pragma unroll
    for (int p = 0; p < 8; ++p) {
        const int idx  = tid + p * 256;    // float4 index, 0..2047
        const int r    = idx >> 4;         // 16 float4 per row
        const int c4   = idx & 15;
        const int grow = rowbase + r;
        v4f v = {0.0f, 0.0f, 0.0f, 0.0f};
        if (grow < n) v = *(const v4f*)(X + (size_t)grow * HDIM + c4 * 4);
        *(v4f*)(sX + r * XPITCH + c4 * 4) = v;
    }
    __syncthreads();

    const int wave = tid >> 5;
    const int lane = tid & 31;
    const int hlf  = lane >> 4;            // which 16-lane half
    const int m    = lane & 15;
    const int rloc = wave * 16 + m;        // local row for A fragment

    // Per-lane base pointers (loop-invariant parts hoisted).
    const float* aptr = sX + rloc * XPITCH + 2 * hlf;
    const float* bptr = sWp + hlf * WPITCH + m * 2;

    v8f acc0 = {}, acc1 = {}, acc2 = {}, acc3 = {};

    #pragma unroll
    for (int k0 = 0; k0 < HDIM; k0 += 4) {
        // A 16x4 f32: lanes 0-15 -> K={k0,k0+1}, lanes 16-31 -> K={k0+2,k0+3}
        v2f a = *(const v2f*)(aptr + k0);
        // B 4x16 f32 per col-tile: single b64 load each, pre-packed K-pairs.
        const float* bk = bptr + (k0 >> 1) * WPITCH;
        v2f b0 = *(const v2f*)(bk +  0);
        v2f b1 = *(const v2f*)(bk + 32);
        v2f b2 = *(const v2f*)(bk + 64);
        v2f b3 = *(const v2f*)(bk + 96);
        acc0 = __builtin_amdgcn_wmma_f32_16x16x4_f32(false, a, false, b0, (short)0, acc0, false, false);
        acc1 = __builtin_amdgcn_wmma_f32_16x16x4_f32(false, a, false, b1, (short)0, acc1, false, false);
        acc2 = __builtin_amdgcn_wmma_f32_16x16x4_f32(false, a, false, b2, (short)0, acc2, false, false);
        acc3 = __builtin_amdgcn_wmma_f32_16x16x4_f32(false, a, false, b3, (short)0, acc3, false, false);
    }

    // Epilogue: per-lane column c_t = 16t + m; fuse bias + relu + wfc2 dot.
    float bc[4], wc[4];
    #pragma unroll
    for (int t = 0; t < 4; ++t) { bc[t] = sB[t * 16 + m]; wc[t] = sW2[t * 16 + m]; }

    float s[8];
    #pragma unroll
    for (int R = 0; R < 8; ++R) {
        float v0 = acc0[R] + bc[0]; v0 = v0 > 0.0f ? v0 : 0.0f;
        float v1 = acc1[R] + bc[1]; v1 = v1 > 0.0f ? v1 : 0.0f;
        float v2 = acc2[R] + bc[2]; v2 = v2 > 0.0f ? v2 : 0.0f;
        float v3 = acc3[R] + bc[3]; v3 = v3 > 0.0f ? v3 : 0.0f;
        s[R] = v0 * wc[0] + v1 * wc[1] + v2 * wc[2] + v3 * wc[3];
    }

    // Reduce across the 16 lanes of each half (masks 1..8 keep halves apart).
    #pragma unroll
    for (int off = 8; off > 0; off >>= 1) {
        #pragma unroll
        for (int R = 0; R < 8; ++R) s[R] += __shfl_xor(s[R], off, 32);
    }

    // C/D layout: lanes 0-15 hold rows 0-7 (VGPR R -> row R), lanes 16-31
    // hold rows 8-15. Lane m==0 of each half writes its 8 row results.
    if (m == 0) {
        const float bias2 = bfc2[0];
        const int r0 = rowbase + wave * 16 + hlf * 8;
        #pragma unroll
        for (int R = 0; R < 8; ++R) {
            const int r = r0 + R;
            if (r < n) out[r] = s[R] + bias2;
        }
    }
}

// ---------------------------------------------------------------------------
extern "C" void kernel_launch(void* const* d_in, const int* in_sizes, int n_in,
                              void* d_out, int out_size, void* d_ws, size_t ws_size,
                              hipStream_t stream) {
    (void)n_in; (void)out_size; (void)ws_size;
    const float* X     = (const float*)d_in[0];   // x_drivers [N,64]
    const float* wproj = (const float*)d_in[9];   // [2,64,64]
    const float* bproj = (const float*)d_in[10];  // [2,64]
    const float* wfc1  = (const float*)d_in[11];  // [128,64]
    const float* bfc1  = (const float*)d_in[12];  // [64]
    const float* wfc2  = (const float*)d_in[13];  // [64,1]
    const float* bfc2  = (const float*)d_in[14];  // [1]
    float* out = (float*)d_out;
    float* ws  = (float*)d_ws;                    // needs (4096+64)*4 bytes

    const int n = in_sizes[0] / HDIM;             // N = 500000

    build_combined_weights<<<1, 256, 0, stream>>>(wproj, bproj, wfc1, bfc1, ws);

    const int blocks = (n + ROWS - 1) / ROWS;
    mpsgnn_fused_gemm<<<blocks, 256, 0, stream>>>(X, ws, ws + 4096, wfc2, bfc2,
                                                  out, n);
}